// UNet_10814727651711
// MI455X (gfx1250) — compile-verified
//
#include <hip/hip_runtime.h>
#include <math.h>
#include <stdint.h>

typedef __attribute__((ext_vector_type(2))) float v2f;
typedef __attribute__((ext_vector_type(8))) float v8f;

static __device__ __forceinline__ v8f wmma4(v2f a, v2f b, v8f c) {
  // D(16x16,f32) += A(16x4,f32) * B(4x16,f32)
  return __builtin_amdgcn_wmma_f32_16x16x4_f32(false, a, false, b, (short)0, c, false, false);
}

// CDNA5 async global->LDS copy (ASYNCcnt-tracked, bypasses VGPRs).
// dst: LDS byte address (low 32 bits of flat shared pointer), 16B aligned; src: 16B aligned global.
static __device__ __forceinline__ void async_b128(void* lds, const void* g) {
  asm volatile("global_load_async_to_lds_b128 %0, %1, off"
               :: "v"((uint32_t)(size_t)lds), "v"(g) : "memory");
}
static __device__ __forceinline__ void async_wait0() {
  asm volatile("s_wait_asynccnt 0" ::: "memory");
}

#define IMGSZ 512
#define NPB   (IMGSZ * IMGSZ)       // tokens per batch = 262144
#define BSZ   2
#define CHK   128
#define NC    (NPB / CHK)           // 2048 chunks per batch
#define GRP   32                    // chunks per scan group
#define NGRP  (NC / GRP)            // 64 groups per batch

// ---------------------------------------------------------------- patch embed
__global__ void patch_embed_k(const float* __restrict__ x, const float* __restrict__ pw,
                              const float* __restrict__ pb, float* __restrict__ h) {
  size_t gid = (size_t)blockIdx.x * blockDim.x + threadIdx.x;   // Ntok*32 threads
  size_t tok = gid >> 5;
  int d = (int)(gid & 31);
  size_t b = tok >> 18;                 // NPB = 2^18
  size_t n = tok & (size_t)(NPB - 1);
  const size_t P = (size_t)NPB;
  float v = pb[d];
  v += x[b * 3 * P + 0 * P + n] * pw[0 * 32 + d];
  v += x[b * 3 * P + 1 * P + n] * pw[1 * 32 + d];
  v += x[b * 3 * P + 2 * P + n] * pw[2 * 32 + d];
  h[tok * 32 + d] = v;
}

// ---------------------------------------------------------------- layernorm D=32
__global__ void ln32_k(const float* __restrict__ X, const float* __restrict__ g,
                       const float* __restrict__ b, float* __restrict__ Y) {
  size_t token = (size_t)blockIdx.x * 8 + (threadIdx.x >> 5);
  int lane = threadIdx.x & 31;
  float xv = X[token * 32 + lane];
  float s = xv;
  for (int o = 16; o; o >>= 1) s += __shfl_xor(s, o, 32);
  float m = s * (1.f / 32.f);
  float d = xv - m;
  float v = d * d;
  for (int o = 16; o; o >>= 1) v += __shfl_xor(v, o, 32);
  v *= (1.f / 32.f);
  Y[token * 32 + lane] = d * rsqrtf(v + 1e-5f) * g[lane] + b[lane];
}

// ---------------------------------------------------------------- generic row GEMM (WMMA)
// Y[row,:] = act(X[row,:Kin] @ W[Kin,Kout] + bias) (+ resid). 64 rows/block, 128 thr (4 waves).
template <int Kin, int Kout, int ACT>
__global__ void gemm_rows_k(const float* __restrict__ X, const float* __restrict__ W,
                            const float* __restrict__ bias, const float* __restrict__ resid,
                            float* __restrict__ Y) {
  constexpr int LDA = Kin + 4;               // *4B is a multiple of 16; banks conflict-free
  __shared__ __align__(16) float As[64 * LDA];
  __shared__ __align__(16) float Ws[Kin * Kout];
  size_t row0 = (size_t)blockIdx.x * 64;

  constexpr int CPR = Kin >> 2;              // 16B chunks per row
  for (int idx = threadIdx.x; idx < 64 * CPR; idx += blockDim.x) {
    int r = idx / CPR, c4 = (idx - r * CPR) * 4;
    async_b128(&As[r * LDA + c4], &X[(row0 + r) * (size_t)Kin + c4]);
  }
  for (int idx = threadIdx.x; idx < (Kin * Kout) >> 2; idx += blockDim.x) {
    async_b128(&Ws[idx * 4], &W[idx * 4]);
  }
  async_wait0();
  __syncthreads();

  int wave = threadIdx.x >> 5, lane = threadIdx.x & 31;
  int n = lane & 15, hi = lane >> 4;
  constexpr int NTN = Kout >> 4;
  for (int t = wave; t < 4 * NTN; t += 4) {
    int tm = t / NTN, tn = t - tm * NTN;
    v8f acc = {};
    const float* Ap = &As[tm * 16 * LDA];
    const float* Bp = &Ws[tn * 16];
    for (int k0 = 0; k0 < Kin; k0 += 4) {
      int kk = k0 + 2 * hi;
      v2f a, bb;
      a.x = Ap[n * LDA + kk];
      a.y = Ap[n * LDA + kk + 1];
      bb.x = Bp[kk * Kout + n];
      bb.y = Bp[(kk + 1) * Kout + n];
      acc = wmma4(a, bb, acc);
    }
#pragma unroll
    for (int r = 0; r < 8; r++) {
      int mm = r + 8 * hi;
      size_t orow = row0 + (size_t)(tm * 16 + mm);
      float v = acc[r] + bias[tn * 16 + n];
      if (ACT) v = 0.5f * v * (1.f + erff(v * 0.70710678118654752f));
      size_t oi = orow * (size_t)Kout + tn * 16 + n;
      if (resid) v += resid[oi];
      Y[oi] = v;
    }
  }
}

// ---------------------------------------------------------------- intra-chunk attention + outer product
// One block per chunk (128 tokens), 128 threads (4 waves).
// o[chunk] = scale * (decay-masked QK^T) V ; Abuf[chunk] = (K .* kdec)^T V  (32x32)
__global__ void attn_intra_k(const float* __restrict__ qkv, float* __restrict__ o,
                             float* __restrict__ Abuf, float s, float scale) {
  const int LD = 36;                         // 144B rows: 16B-aligned, conflict-free columns
  __shared__ __align__(16) float Qs[CHK * LD];
  __shared__ __align__(16) float Ks[CHK * LD];
  __shared__ __align__(16) float Vs[CHK * LD];
  __shared__ float Ssc[4][16 * 17];

  size_t chunk = blockIdx.x;
  size_t n0 = chunk * CHK;
  for (int idx = threadIdx.x; idx < CHK * 8; idx += blockDim.x) {   // 8 x 16B per token
    int t = idx >> 3, c4 = (idx & 7) * 4;
    const float* row = &qkv[(n0 + t) * 96];
    async_b128(&Qs[t * LD + c4], row + c4);
    async_b128(&Ks[t * LD + c4], row + 32 + c4);
    async_b128(&Vs[t * LD + c4], row + 64 + c4);
  }
  async_wait0();
  __syncthreads();

  int wave = threadIdx.x >> 5, lane = threadIdx.x & 31;
  int n = lane & 15, hi = lane >> 4;

  // 16 output tiles (8 row-tiles x 2 col-tiles), 4 per wave; uniform loop shape across waves.
  for (int ot = 0; ot < 4; ot++) {
    int t = wave * 4 + ot;
    int tm = t >> 1, tn = t & 1;
    v8f acc = {};
    for (int jt = 0; jt < 8; jt++) {
      v8f sacc = {};
      for (int k0 = 0; k0 < 32; k0 += 4) {
        int kk = k0 + 2 * hi;
        v2f a, bb;
        a.x = Qs[(tm * 16 + n) * LD + kk];
        a.y = Qs[(tm * 16 + n) * LD + kk + 1];
        bb.x = Ks[(jt * 16 + n) * LD + kk];    // B = K^T
        bb.y = Ks[(jt * 16 + n) * LD + kk + 1];
        sacc = wmma4(a, bb, sacc);
      }
      // scale + decay mask, stage S tile (row-major) in per-wave scratch
#pragma unroll
      for (int r = 0; r < 8; r++) {
        int i = tm * 16 + r + 8 * hi;
        int j = jt * 16 + n;
        float v = (i >= j) ? sacc[r] * (scale * __expf(-s * (float)(i - j))) : 0.f;
        Ssc[wave][(r + 8 * hi) * 17 + n] = v;
      }
      __syncthreads();
      for (int k0 = 0; k0 < 16; k0 += 4) {
        int kk = k0 + 2 * hi;
        v2f a, bb;
        a.x = Ssc[wave][n * 17 + kk];
        a.y = Ssc[wave][n * 17 + kk + 1];
        bb.x = Vs[(jt * 16 + kk) * LD + tn * 16 + n];
        bb.y = Vs[(jt * 16 + kk + 1) * LD + tn * 16 + n];
        acc = wmma4(a, bb, acc);
      }
      __syncthreads();
    }
#pragma unroll
    for (int r = 0; r < 8; r++) {
      int i = tm * 16 + r + 8 * hi;
      o[(n0 + i) * 32 + tn * 16 + n] = acc[r];
    }
  }

  // per-chunk kv contribution A = (K .* kdec)^T V : 4 16x16 tiles, one per wave
  {
    int tm = wave >> 1, tn = wave & 1;
    v8f acc = {};
    for (int k0 = 0; k0 < CHK; k0 += 4) {
      int kk = k0 + 2 * hi;
      float kd0 = __expf(-s * (float)(CHK - 1 - kk));
      float kd1 = __expf(-s * (float)(CHK - 2 - kk));
      v2f a, bb;
      a.x = Ks[kk * LD + tm * 16 + n] * kd0;         // A[d][j] = K[j][d]*kdec[j]
      a.y = Ks[(kk + 1) * LD + tm * 16 + n] * kd1;
      bb.x = Vs[kk * LD + tn * 16 + n];
      bb.y = Vs[(kk + 1) * LD + tn * 16 + n];
      acc = wmma4(a, bb, acc);
    }
#pragma unroll
    for (int r = 0; r < 8; r++) {
      int d = tm * 16 + r + 8 * hi;
      Abuf[chunk * 1024 + d * 32 + tn * 16 + n] = acc[r];
    }
  }
}

// ---------------------------------------------------------------- kv decay scan, pass 1
__global__ void kv_scan1_k(float* __restrict__ L, float* __restrict__ Sg, float dec) {
  size_t grp = blockIdx.x;
  int e0 = threadIdx.x * 4;
  float r0 = 0.f, r1 = 0.f, r2 = 0.f, r3 = 0.f;
  size_t base = grp * GRP * 1024;
  for (int i = 0; i < GRP; i++) {
    size_t idx = base + (size_t)i * 1024 + e0;
    __builtin_prefetch(&L[idx + 1024], 0, 1);      // global_prefetch for next chunk
    float a0 = L[idx], a1 = L[idx + 1], a2 = L[idx + 2], a3 = L[idx + 3];
    L[idx] = r0; L[idx + 1] = r1; L[idx + 2] = r2; L[idx + 3] = r3;
    r0 = dec * r0 + a0; r1 = dec * r1 + a1; r2 = dec * r2 + a2; r3 = dec * r3 + a3;
  }
  size_t sidx = grp * 1024 + e0;
  Sg[sidx] = r0; Sg[sidx + 1] = r1; Sg[sidx + 2] = r2; Sg[sidx + 3] = r3;
}

// pass 2: exclusive scan over groups per batch. grid = B, 256 threads x 4 elems.
__global__ void kv_scan2_k(const float* __restrict__ Sg, float* __restrict__ Cg, float decG) {
  size_t b = blockIdx.x;
  int e0 = threadIdx.x * 4;
  float c0 = 0.f, c1 = 0.f, c2 = 0.f, c3 = 0.f;
  for (int g = 0; g < NGRP; g++) {
    size_t idx = (b * NGRP + g) * 1024 + e0;
    Cg[idx] = c0; Cg[idx + 1] = c1; Cg[idx + 2] = c2; Cg[idx + 3] = c3;
    c0 = decG * c0 + Sg[idx]; c1 = decG * c1 + Sg[idx + 1];
    c2 = decG * c2 + Sg[idx + 2]; c3 = decG * c3 + Sg[idx + 3];
  }
}

// ---------------------------------------------------------------- inter-chunk attention
// o[chunk] += diag(scale*qdec) Q @ kv_state,  kv_state = dec^iloc * Cg + L
__global__ void attn_inter_k(const float* __restrict__ qkv, const float* __restrict__ L,
                             const float* __restrict__ Cg, float* __restrict__ o,
                             float s, float scale) {
  const int LD = 36;
  __shared__ __align__(16) float Qs[CHK * LD];
  __shared__ float Es[32 * 33];
  size_t chunk = blockIdx.x;
  size_t b = chunk / NC;
  int ci = (int)(chunk - b * NC);
  int g = ci >> 5, iloc = ci & 31;
  float decp = __expf(-s * (float)CHK * (float)iloc);
  size_t n0 = chunk * CHK;

  for (int idx = threadIdx.x; idx < 1024; idx += blockDim.x) {
    float E = decp * Cg[(b * NGRP + g) * 1024 + idx] + L[chunk * 1024 + idx];
    int d = idx >> 5, e = idx & 31;
    Es[d * 33 + e] = E;
  }
  for (int idx = threadIdx.x; idx < CHK * 8; idx += blockDim.x) {   // raw async Q stage
    int t = idx >> 3, c4 = (idx & 7) * 4;
    async_b128(&Qs[t * LD + c4], &qkv[(n0 + t) * 96 + c4]);
  }
  async_wait0();
  __syncthreads();

  int wave = threadIdx.x >> 5, lane = threadIdx.x & 31;
  int n = lane & 15, hi = lane >> 4;
  for (int ot = 0; ot < 4; ot++) {
    int t = wave * 4 + ot;
    int tm = t >> 1, tn = t & 1;
    v8f acc = {};
    for (int k0 = 0; k0 < 32; k0 += 4) {
      int kk = k0 + 2 * hi;
      v2f a, bb;
      a.x = Qs[(tm * 16 + n) * LD + kk];
      a.y = Qs[(tm * 16 + n) * LD + kk + 1];
      bb.x = Es[kk * 33 + tn * 16 + n];
      bb.y = Es[(kk + 1) * 33 + tn * 16 + n];
      acc = wmma4(a, bb, acc);
    }
#pragma unroll
    for (int r = 0; r < 8; r++) {
      int i = tm * 16 + r + 8 * hi;
      float fac = scale * __expf(-s * (float)(i + 1));   // row factor folded out of staging
      size_t oi = (n0 + i) * 32 + tn * 16 + n;
      o[oi] = o[oi] + acc[r] * fac;
    }
  }
}

// ---------------------------------------------------------------- 2x2 merge + LN(128) + GEMM(128x64)
// 32 output rows per block, 256 threads (8 waves).
__global__ void merge_ln_gemm_k(const float* __restrict__ h, const float* __restrict__ g,
                                const float* __restrict__ bb, const float* __restrict__ W,
                                float* __restrict__ out) {
  const int LD = 132;
  __shared__ __align__(16) float As[32 * 132];   // 16.9 KB
  __shared__ __align__(16) float Ws[128 * 64];   // 32 KB
  const int HW = IMGSZ / 2;        // 256
  const int Nout = HW * HW;        // 65536 per batch
  size_t row0 = (size_t)blockIdx.x * 32;

  for (int idx = threadIdx.x; idx < 32 * 32; idx += blockDim.x) {   // 16B chunks
    int r = idx >> 5, c = (idx & 31) * 4;
    size_t tok = row0 + r;
    size_t b = tok / Nout;
    int t = (int)(tok - b * Nout);
    int ro = t / HW, co = t - ro * HW;
    int q = c >> 5, ch = c & 31;
    int dr = q & 1, dc = q >> 1;                 // [x0,x1,x2,x3] = (0,0),(1,0),(0,1),(1,1)
    size_t n = b * (size_t)NPB + (size_t)(2 * ro + dr) * IMGSZ + (2 * co + dc);
    async_b128(&As[r * LD + c], &h[n * 32 + ch]);
  }
  for (int idx = threadIdx.x; idx < (128 * 64) >> 2; idx += blockDim.x) {
    async_b128(&Ws[idx * 4], &W[idx * 4]);
  }
  async_wait0();
  __syncthreads();

  // LN over 128 channels: 8 lanes per row (rows aligned within waves)
  {
    int r = threadIdx.x >> 3, part = threadIdx.x & 7;
    float sm = 0.f, sq = 0.f;
    for (int c = part * 16; c < part * 16 + 16; c++) {
      float v = As[r * LD + c];
      sm += v; sq += v * v;
    }
    for (int o = 1; o < 8; o <<= 1) { sm += __shfl_xor(sm, o, 32); sq += __shfl_xor(sq, o, 32); }
    float m = sm * (1.f / 128.f);
    float var = sq * (1.f / 128.f) - m * m;
    float rs = rsqrtf(var + 1e-5f);
    for (int c = part * 16; c < part * 16 + 16; c++) {
      As[r * LD + c] = (As[r * LD + c] - m) * rs * g[c] + bb[c];
    }
  }
  __syncthreads();

  // GEMM 32x128 @ 128x64 : 2x4 = 8 tiles, one per wave
  int wave = threadIdx.x >> 5, lane = threadIdx.x & 31;
  int n = lane & 15, hi = lane >> 4;
  int tm = wave >> 2, tn = wave & 3;
  v8f acc = {};
  for (int k0 = 0; k0 < 128; k0 += 4) {
    int kk = k0 + 2 * hi;
    v2f a, bv;
    a.x = As[(tm * 16 + n) * LD + kk];
    a.y = As[(tm * 16 + n) * LD + kk + 1];
    bv.x = Ws[kk * 64 + tn * 16 + n];
    bv.y = Ws[(kk + 1) * 64 + tn * 16 + n];
    acc = wmma4(a, bv, acc);
  }
#pragma unroll
  for (int r = 0; r < 8; r++) {
    size_t orow = row0 + (size_t)(tm * 16 + r + 8 * hi);
    out[orow * 64 + tn * 16 + n] = acc[r];
  }
}

// ================================================================ host launch
extern "C" void kernel_launch(void* const* d_in, const int* in_sizes, int n_in,
                              void* d_out, int out_size, void* d_ws, size_t ws_size,
                              hipStream_t stream) {
  (void)in_sizes; (void)n_in; (void)out_size; (void)ws_size;
  const float* x       = (const float*)d_in[0];
  const float* patch_w = (const float*)d_in[1];
  const float* patch_b = (const float*)d_in[2];
  const float* ln1_g   = (const float*)d_in[3];
  const float* ln1_b   = (const float*)d_in[4];
  const float* qkv_w   = (const float*)d_in[5];
  const float* qkv_b   = (const float*)d_in[6];
  const float* proj_w  = (const float*)d_in[7];
  const float* proj_b  = (const float*)d_in[8];
  const float* ln2_g   = (const float*)d_in[9];
  const float* ln2_b   = (const float*)d_in[10];
  const float* fc1_w   = (const float*)d_in[11];
  const float* fc1_b   = (const float*)d_in[12];
  const float* fc2_w   = (const float*)d_in[13];
  const float* fc2_b   = (const float*)d_in[14];
  const float* merge_g = (const float*)d_in[15];
  const float* merge_b = (const float*)d_in[16];
  const float* merge_w = (const float*)d_in[17];

  const size_t Ntok = (size_t)BSZ * NPB;           // 524288
  const float s     = 0.00390625f;                 // build_slopes(1)
  const float scale = 0.17677669529663689f;        // 32^-0.5
  const float dec   = expf(-s * (float)CHK);       // exp(-0.5)
  const float decG  = expf(-s * (float)CHK * (float)GRP);

  char* w = (char*)d_ws;
  float* h   = (float*)w; w += Ntok * 32 * sizeof(float);
  float* y   = (float*)w; w += Ntok * 32 * sizeof(float);
  float* big = (float*)w; w += Ntok * 128 * sizeof(float);  // qkv (96) then MLP hidden (128)
  float* o   = (float*)w; w += Ntok * 32 * sizeof(float);
  float* L   = (float*)w; w += (size_t)BSZ * NC * 1024 * sizeof(float);
  float* Sg  = (float*)w; w += (size_t)BSZ * NGRP * 1024 * sizeof(float);
  float* Cg  = (float*)w; w += (size_t)BSZ * NGRP * 1024 * sizeof(float);

  patch_embed_k<<<(unsigned)(Ntok * 32 / 256), 256, 0, stream>>>(x, patch_w, patch_b, h);

  for (int l = 0; l < 2; l++) {
    ln32_k<<<(unsigned)(Ntok / 8), 256, 0, stream>>>(h, ln1_g + l * 32, ln1_b + l * 32, y);
    gemm_rows_k<32, 96, 0><<<(unsigned)(Ntok / 64), 128, 0, stream>>>(
        y, qkv_w + (size_t)l * 32 * 96, qkv_b + l * 96, nullptr, big);
    attn_intra_k<<<(unsigned)(BSZ * NC), 128, 0, stream>>>(big, o, L, s, scale);
    kv_scan1_k<<<(unsigned)(BSZ * NGRP), 256, 0, stream>>>(L, Sg, dec);
    kv_scan2_k<<<(unsigned)BSZ, 256, 0, stream>>>(Sg, Cg, decG);
    attn_inter_k<<<(unsigned)(BSZ * NC), 128, 0, stream>>>(big, L, Cg, o, s, scale);
    gemm_rows_k<32, 32, 0><<<(unsigned)(Ntok / 64), 128, 0, stream>>>(
        o, proj_w + (size_t)l * 32 * 32, proj_b + l * 32, h, h);
    ln32_k<<<(unsigned)(Ntok / 8), 256, 0, stream>>>(h, ln2_g + l * 32, ln2_b + l * 32, y);
    gemm_rows_k<32, 128, 1><<<(unsigned)(Ntok / 64), 128, 0, stream>>>(
        y, fc1_w + (size_t)l * 32 * 128, fc1_b + l * 128, nullptr, big);
    gemm_rows_k<128, 32, 0><<<(unsigned)(Ntok / 64), 128, 0, stream>>>(
        big, fc2_w + (size_t)l * 128 * 32, fc2_b + l * 32, h, h);
  }

  merge_ln_gemm_k<<<(unsigned)(BSZ * (NPB / 4) / 32), 256, 0, stream>>>(
      h, merge_g, merge_b, merge_w, (float*)d_out);
}